// ImprovedDilatedAttention_61727269978794
// MI455X (gfx1250) — compile-verified
//
#include <hip/hip_runtime.h>
#include <math.h>

typedef float v2f __attribute__((ext_vector_type(2)));
typedef float v8f __attribute__((ext_vector_type(8)));

#define B_DIM 2
#define N_DIM 8192
#define H_DIM 16
#define D_DIM 64
#define WAVES_PER_BLOCK 8
#define NPOS_TOTAL (B_DIM * N_DIM)

__device__ __forceinline__ int head_group(int h) {
    // heads 0..5 -> group 0, 6..10 -> group 1, 11..15 -> group 2
    return (h < 6) ? 0 : ((h < 11) ? 1 : 2);
}

__global__ __launch_bounds__(256) void dilated_headmix_attn_kernel(
    const float* __restrict__ q,
    const float* __restrict__ k,
    const float* __restrict__ v,
    float* __restrict__ out)
{
    __shared__ float ldsA[WAVES_PER_BLOCK][16 * 16];

    const int tid  = threadIdx.x;
    const int wave = tid >> 5;
    const int lane = tid & 31;
    const int lo   = lane & 15;   // head row (A/B loads) / column N (C/D layout)
    const int hi   = lane >> 4;   // which K-pair within a 4-wide chunk

    // one wave per (b, pos); w == b*N + pos
    const int w   = blockIdx.x * WAVES_PER_BLOCK + wave;
    const int pos = w & (N_DIM - 1);

    const size_t base = (size_t)w * (H_DIM * D_DIM);

    // ---------------- S = Q(16x64) @ K^T(64x16) ----------------
    // f32 WMMA A(16x4)/B(4x16) layouts coincide as (head, d) maps:
    //   lane = head + 16*(k>=2), vgpr = k&1  ->  aligned float2 per chunk.
    v2f aq[16], bk[16];
#pragma unroll
    for (int c = 0; c < 16; ++c) {
        const int d0 = 4 * c + 2 * hi;
        aq[c] = *(const v2f*)(q + base + (size_t)lo * D_DIM + d0);
        bk[c] = *(const v2f*)(k + base + (size_t)lo * D_DIM + d0);
    }
    v8f acc0 = {};
    v8f acc1 = {};
#pragma unroll
    for (int c = 0; c < 16; c += 2) {
        acc0 = __builtin_amdgcn_wmma_f32_16x16x4_f32(
            false, aq[c],     false, bk[c],     (short)0, acc0, false, false);
        acc1 = __builtin_amdgcn_wmma_f32_16x16x4_f32(
            false, aq[c + 1], false, bk[c + 1], (short)0, acc1, false, false);
    }

    // ---------------- masked row softmax over 16 key heads ----------------
    const float scale = 0.125f;           // 1/sqrt(64)
    const int   gn    = head_group(lo);   // key-head group of this column
    float arow[8];
#pragma unroll
    for (int vv = 0; vv < 8; ++vv) {
        const int m = vv + 8 * hi;        // query head of this C/D row
        float x = (acc0[vv] + acc1[vv]) * scale;
        if (head_group(m) != gn) x = -__builtin_inff();  // cross-group mask
        float rmax = x;
#pragma unroll
        for (int s = 1; s < 16; s <<= 1)
            rmax = fmaxf(rmax, __shfl_xor(rmax, s, 32));
        const float e = __expf(x - rmax);
        float rsum = e;
#pragma unroll
        for (int s = 1; s < 16; s <<= 1)
            rsum += __shfl_xor(rsum, s, 32);
        arow[vv] = e / rsum;
    }

    // ---------------- transpose A: C/D layout -> A layout via LDS ----------------
#pragma unroll
    for (int vv = 0; vv < 8; ++vv) {
        const int m = vv + 8 * hi;
        ldsA[wave][m * 16 + lo] = arow[vv];
    }
    __syncthreads();

    v2f av[4];
#pragma unroll
    for (int kc = 0; kc < 4; ++kc)
        av[kc] = *(const v2f*)&ldsA[wave][lo * 16 + 4 * kc + 2 * hi];

    // ---------------- O = A(16x16) @ V(16x64) ----------------
    v8f o[4];
#pragma unroll
    for (int t = 0; t < 4; ++t) {
        v8f oc = {};
#pragma unroll
        for (int kc = 0; kc < 4; ++kc) {
            const int h0 = 4 * kc + 2 * hi;  // key head supplying this B pair
            v2f vb;
            vb.x = v[base + (size_t)h0 * D_DIM       + 16 * t + lo];
            vb.y = v[base + (size_t)(h0 + 1) * D_DIM + 16 * t + lo];
            oc = __builtin_amdgcn_wmma_f32_16x16x4_f32(
                false, av[kc], false, vb, (short)0, oc, false, false);
        }
        o[t] = oc;
    }

    // ---------------- store with dilation activity mask, /NUM_GROUPS ----------------
    const bool act1 = ((pos & 1) == 1);   // group 1: r=2, offset 1
    const bool act2 = ((pos & 3) == 2);   // group 2: r=4, offset 2
    const float inv3 = 1.0f / 3.0f;
#pragma unroll
    for (int t = 0; t < 4; ++t) {
#pragma unroll
        for (int vv = 0; vv < 8; ++vv) {
            const int m = vv + 8 * hi;
            const int g = head_group(m);
            const bool active = (g == 0) | ((g == 1) & act1) | ((g == 2) & act2);
            const float val = active ? (o[t][vv] * inv3) : 0.0f;
            out[base + (size_t)m * D_DIM + 16 * t + lo] = val;
        }
    }
}

extern "C" void kernel_launch(void* const* d_in, const int* in_sizes, int n_in,
                              void* d_out, int out_size, void* d_ws, size_t ws_size,
                              hipStream_t stream) {
    const float* q = (const float*)d_in[0];
    const float* k = (const float*)d_in[1];
    const float* v = (const float*)d_in[2];
    float* out = (float*)d_out;

    const int blocks = NPOS_TOTAL / WAVES_PER_BLOCK;  // 16384 waves / 8 = 2048 blocks
    dilated_headmix_attn_kernel<<<blocks, 256, 0, stream>>>(q, k, v, out);
}